// BootstrappedCrossEntropyLoss_64287070486991
// MI455X (gfx1250) — compile-verified
//
#include <hip/hip_runtime.h>
#include <math.h>
#include <stdint.h>

#define BATCH   16
#define NPB     (1u << 20)   // elements per batch (C*H*W)
#define KSEL    65536u       // top-K per batch
#define HBLK    64           // blocks per batch for streaming passes
#define THREADS 256

// ---- workspace layout in 32-bit words ----
#define OFF_H0   0
#define OFF_H1   (OFF_H0 + BATCH*4096)
#define OFF_H2   (OFF_H1 + BATCH*4096)
#define OFF_S0   (OFF_H2 + BATCH*256)
#define OFF_S1   (OFF_S0 + BATCH*2)
#define OFF_S2   (OFF_S1 + BATCH*2)
#define OFF_PART (OFF_S2 + BATCH*2)
#define WS_WORDS (OFF_PART + BATCH*HBLK)

typedef __attribute__((ext_vector_type(2))) float v2f;
typedef __attribute__((ext_vector_type(8))) float v8f;

// Stable BCE-with-logits cross-entropy per pixel (non-negative result).
__device__ __forceinline__ float xent(float o, float l) {
    float lab = (l >= 0.5f) ? 1.0f : 0.0f;
    float pos = (o >= 0.0f) ? 1.0f : 0.0f;
    return o * (pos - lab) + log1pf(expf(-fabsf(o)));
}

__global__ void zero_ws(uint32_t* ws, int nwords) {
    int i = blockIdx.x * blockDim.x + threadIdx.x;
    int stride = gridDim.x * blockDim.x;
    for (; i < nwords; i += stride) ws[i] = 0u;
}

// Histogram pass. level 0: bin = bits>>20 (4096 bins), optionally writes CE cache.
// level 1: filter (bits>>20)==prefix, bin=(bits>>8)&0xFFF.
// level 2: filter (bits>>8)==prefix,  bin=bits&0xFF (256 bins).
__global__ void __launch_bounds__(THREADS)
hist_pass(const float* __restrict__ outp, const float* __restrict__ labp,
          const float* __restrict__ ce_cache, float* __restrict__ ce_store,
          uint32_t* __restrict__ ghist, const uint32_t* __restrict__ prevState,
          int level, int nbins)
{
    __shared__ uint32_t lh[4096];
    const int batch = blockIdx.y;
    for (int i = threadIdx.x; i < nbins; i += blockDim.x) lh[i] = 0u;
    uint32_t prefix = 0u;
    if (level > 0) prefix = prevState[batch * 2];
    __syncthreads();

    const size_t base = (size_t)batch * NPB;
    const int nvec = NPB / 4;
    const float4* o4 = (const float4*)(outp + base);
    const float4* l4 = (const float4*)(labp + base);
    const float4* c4 = ce_cache ? (const float4*)(ce_cache + base) : nullptr;
    float4*       s4 = ce_store ? (float4*)(ce_store + base) : nullptr;

    for (int v = blockIdx.x * blockDim.x + threadIdx.x; v < nvec;
         v += gridDim.x * blockDim.x) {
        float4 ce;
        if (c4) {
            ce = c4[v];
        } else {
            float4 o = o4[v];
            float4 l = l4[v];
            ce.x = xent(o.x, l.x); ce.y = xent(o.y, l.y);
            ce.z = xent(o.z, l.z); ce.w = xent(o.w, l.w);
            if (s4) s4[v] = ce;   // keep CE resident in 192MB L2 for later passes
        }
        float cearr[4] = {ce.x, ce.y, ce.z, ce.w};
#pragma unroll
        for (int j = 0; j < 4; j++) {
            uint32_t b = __float_as_uint(cearr[j]);  // non-negative -> uint order
            if (level == 0) {
                atomicAdd(&lh[b >> 20], 1u);
            } else if (level == 1) {
                if ((b >> 20) == prefix) atomicAdd(&lh[(b >> 8) & 0xFFFu], 1u);
            } else {
                if ((b >> 8) == prefix) atomicAdd(&lh[b & 0xFFu], 1u);
            }
        }
    }
    __syncthreads();
    uint32_t* gh = ghist + (size_t)batch * nbins;
    for (int i = threadIdx.x; i < nbins; i += blockDim.x)
        if (lh[i]) atomicAdd(&gh[i], lh[i]);
}

// One block per batch: scan histogram from the top, find crossing bin + residual need.
__global__ void select_pass(const uint32_t* __restrict__ ghist, int nbins,
                            const uint32_t* __restrict__ prevState,
                            uint32_t* __restrict__ outState, int level)
{
    __shared__ uint32_t h[4096];
    __shared__ uint32_t csum[256];
    const int batch = blockIdx.x;
    const uint32_t* gh = ghist + (size_t)batch * nbins;
    const int chunks = nbins >> 4;
    if ((int)threadIdx.x < chunks) {
        uint32_t s = 0u;
        const int b0 = threadIdx.x * 16;
        for (int j = 0; j < 16; j++) {
            uint32_t x = gh[b0 + j];
            h[b0 + j] = x;
            s += x;
        }
        csum[threadIdx.x] = s;
    }
    __syncthreads();
    if (threadIdx.x == 0) {
        uint32_t target = (level == 0) ? KSEL : prevState[batch * 2 + 1];
        uint32_t acc = 0u;
        int c = chunks - 1;
        for (; c > 0; c--) {
            if (acc + csum[c] >= target) break;
            acc += csum[c];
        }
        int bsel = c * 16;
        for (int bin = c * 16 + 15; bin >= c * 16; bin--) {
            if (acc + h[bin] >= target) { bsel = bin; break; }
            acc += h[bin];
        }
        uint32_t need = target - acc;   // elements to take from bin bsel (>=1)
        uint32_t px;
        if (level == 0)      px = (uint32_t)bsel;
        else if (level == 1) px = (prevState[batch * 2] << 12) | (uint32_t)bsel;
        else                 px = (prevState[batch * 2] << 8)  | (uint32_t)bsel; // = thr bits
        outState[batch * 2]     = px;
        outState[batch * 2 + 1] = need;
    }
}

// Sum of all CE values strictly above threshold; fixed-order block reduction.
__global__ void __launch_bounds__(THREADS)
sum_pass(const float* __restrict__ outp, const float* __restrict__ labp,
         const float* __restrict__ ce_cache, const uint32_t* __restrict__ state2,
         float* __restrict__ partials)
{
    const int batch = blockIdx.y;
    const uint32_t thr = state2[batch * 2];
    const size_t base = (size_t)batch * NPB;
    const int nvec = NPB / 4;
    const float4* o4 = (const float4*)(outp + base);
    const float4* l4 = (const float4*)(labp + base);
    const float4* c4 = ce_cache ? (const float4*)(ce_cache + base) : nullptr;

    float acc = 0.0f;
    for (int v = blockIdx.x * blockDim.x + threadIdx.x; v < nvec;
         v += gridDim.x * blockDim.x) {
        float4 ce;
        if (c4) {
            ce = c4[v];
        } else {
            float4 o = o4[v];
            float4 l = l4[v];
            ce.x = xent(o.x, l.x); ce.y = xent(o.y, l.y);
            ce.z = xent(o.z, l.z); ce.w = xent(o.w, l.w);
        }
        float cearr[4] = {ce.x, ce.y, ce.z, ce.w};
#pragma unroll
        for (int j = 0; j < 4; j++)
            if (__float_as_uint(cearr[j]) > thr) acc += cearr[j];
    }
    // wave32 reduction
    for (int off = 16; off > 0; off >>= 1) acc += __shfl_xor(acc, off, 32);
    __shared__ float wsum[THREADS / 32];
    const int wave = threadIdx.x >> 5, lane = threadIdx.x & 31;
    if (lane == 0) wsum[wave] = acc;
    __syncthreads();
    if (threadIdx.x == 0) {
        float s = 0.0f;
        for (int w = 0; w < THREADS / 32; w++) s += wsum[w];
        partials[batch * HBLK + blockIdx.x] = s;
    }
}

// 16 waves, one per batch. Each wave reduces its 64 block-partials with a single
// f32 WMMA (A = 16x4 tile of partials, B = ones) -> 16 exact row sums, then one
// cross-half shuffle. Adds tie correction rem*thr, averages over batch.
__global__ void __launch_bounds__(512)
final_pass(const float* __restrict__ partials, const uint32_t* __restrict__ state2,
           float* __restrict__ out)
{
    const int wave = threadIdx.x >> 5;   // == batch index, 0..15
    const int lane = threadIdx.x & 31;
    const float* P = partials + wave * HBLK;
    // 16x4 f32 A layout: lanes 0-15 hold {K0,K1} of row M=lane; lanes 16-31 {K2,K3}.
    const int m  = lane & 15;
    const int kb = (lane >> 4) * 2;
    v2f a; a.x = P[m * 4 + kb]; a.y = P[m * 4 + kb + 1];
    v2f b; b.x = 1.0f; b.y = 1.0f;     // ones matrix: D rows = exact row sums
    v8f c = {};
    c = __builtin_amdgcn_wmma_f32_16x16x4_f32(
        /*neg_a=*/false, a, /*neg_b=*/false, b,
        /*c_mod=*/(short)0, c, /*reuse_a=*/false, /*reuse_b=*/false);
    // lanes 0-15 hold rows 0..7 (all columns equal); lanes 16-31 hold rows 8..15
    float s = c[0] + c[1] + c[2] + c[3] + c[4] + c[5] + c[6] + c[7];
    s += __shfl_xor(s, 16, 32);        // full sum of 64 partials in every lane

    __shared__ float bm[BATCH];
    if (lane == 0) {
        uint32_t thr = state2[wave * 2];
        uint32_t rem = state2[wave * 2 + 1];
        float total = s + (float)rem * __uint_as_float(thr);
        bm[wave] = total / (float)KSEL;
    }
    __syncthreads();
    if (threadIdx.x == 0) {
        float t = 0.0f;
        for (int i = 0; i < BATCH; i++) t += bm[i];
        out[0] = t / (float)BATCH;
    }
}

extern "C" void kernel_launch(void* const* d_in, const int* in_sizes, int n_in,
                              void* d_out, int out_size, void* d_ws, size_t ws_size,
                              hipStream_t stream)
{
    const float* outp = (const float*)d_in[0];
    const float* labp = (const float*)d_in[1];
    uint32_t* ws = (uint32_t*)d_ws;
    float* dout = (float*)d_out;

    uint32_t* h0 = ws + OFF_H0;
    uint32_t* h1 = ws + OFF_H1;
    uint32_t* h2 = ws + OFF_H2;
    uint32_t* s0 = ws + OFF_S0;
    uint32_t* s1 = ws + OFF_S1;
    uint32_t* s2 = ws + OFF_S2;
    float*  part = (float*)(ws + OFF_PART);

    // L2-resident CE cache (64MB) if workspace is big enough.
    const size_t cacheBytes = (size_t)BATCH * NPB * sizeof(float);
    const bool useCache = ws_size >= (size_t)WS_WORDS * 4 + cacheBytes;
    float* ceW = useCache ? (float*)(ws + WS_WORDS) : nullptr;  // store target
    const float* ceR = ceW;                                      // read source

    zero_ws<<<dim3(512), dim3(THREADS), 0, stream>>>(ws, WS_WORDS);

    dim3 sgrid(HBLK, BATCH);
    // Level 0: compute CE from HBM inputs, populate cache + 4096-bin histogram.
    hist_pass<<<sgrid, THREADS, 0, stream>>>(outp, labp, nullptr, ceW, h0, nullptr, 0, 4096);
    select_pass<<<BATCH, 256, 0, stream>>>(h0, 4096, nullptr, s0, 0);
    // Levels 1-2 + sum: read CE cache (L2) when available, else recompute.
    hist_pass<<<sgrid, THREADS, 0, stream>>>(outp, labp, ceR, nullptr, h1, s0, 1, 4096);
    select_pass<<<BATCH, 256, 0, stream>>>(h1, 4096, s0, s1, 1);
    hist_pass<<<sgrid, THREADS, 0, stream>>>(outp, labp, ceR, nullptr, h2, s1, 2, 256);
    select_pass<<<BATCH, 256, 0, stream>>>(h2, 256, s1, s2, 2);
    sum_pass<<<sgrid, THREADS, 0, stream>>>(outp, labp, ceR, s2, part);
    final_pass<<<1, 512, 0, stream>>>(part, s2, dout);
}